// NequipLayer_80401787781524
// MI455X (gfx1250) — compile-verified
//
#include <hip/hip_runtime.h>
#include <hip/hip_bf16.h>

typedef float v2f __attribute__((ext_vector_type(2)));
typedef float v8f __attribute__((ext_vector_type(8)));

#define N_NODES   16384
#define N_EDGES   262144
#define MUL       32

__device__ __forceinline__ float silu_f(float x) {
    return x / (1.0f + __expf(-x));
}

__device__ __forceinline__ int rfl(int x) {
    return __builtin_amdgcn_readfirstlane(x);
}

// ---------------------------------------------------------------------------
// Kernel 1: up-projection.  U[n][0:32]=s1, U[n][32+32i : +32]=v1_i  (planar)
// s1 = s @ W_up_s * 32^-0.5 ; v1_i = v_i @ W_up_v * 32^-0.5
// One wave per (16-node tile, chunk c in {s,v0,v1,v2}, n-half nb).
// Branch-free A addressing: (abase, astride) selected once per wave (scalar).
// ---------------------------------------------------------------------------
__global__ void up_kernel(const float* __restrict__ nf,
                          const float* __restrict__ Wus,
                          const float* __restrict__ Wuv,
                          float* __restrict__ U) {
    const int wave = rfl(threadIdx.x >> 5);   // 0..7  (scalar)
    const int lane = threadIdx.x & 31;
    const int half = lane >> 4;               // selects K pair
    const int lm   = lane & 15;               // row (A) / col (B)
    const int nbase = blockIdx.x * 16;
    const int c  = wave >> 1;                 // 0=s, 1..3 = v component (scalar)
    const int nb = wave & 1;                  // N slice (scalar)

    const float* __restrict__ W = (c == 0) ? Wus : Wuv;   // scalar select
    const int astride = (c == 0) ? 1 : 3;                 // scalar
    const int abase   = (c == 0) ? 0 : (31 + c);          // 32 + (c-1)

    // A[lm][k] = nf[(nbase+lm)*128 + abase + astride*k]
    const float* __restrict__ Ap = nf + (size_t)(nbase + lm) * 128
                                      + abase + astride * 2 * half;
    // B[k][col] = W[k*32 + nb*16 + lm]
    const float* __restrict__ Bp = W + (2 * half) * 32 + nb * 16 + lm;

    v8f acc = {};
    #pragma unroll
    for (int kk = 0; kk < 8; ++kk) {
        v2f a, b;
        a.x = Ap[astride * (4 * kk)];
        a.y = Ap[astride * (4 * kk + 1)];
        b.x = Bp[(4 * kk) * 32];
        b.y = Bp[(4 * kk + 1) * 32];
        acc = __builtin_amdgcn_wmma_f32_16x16x4_f32(false, a, false, b,
                                                    (short)0, acc, false, false);
    }
    const float scale = 0.17677669529663687f;   // 32^-0.5
    const int colbase = c * 32 + nb * 16 + lm;
    #pragma unroll
    for (int r = 0; r < 8; ++r) {
        const int row = nbase + r + 8 * half;
        U[(size_t)row * 128 + colbase] = acc[r] * scale;
    }
}

// ---------------------------------------------------------------------------
// Kernel 2: zero accumulators (vectorized b128 stores).
// ---------------------------------------------------------------------------
__global__ void zero_kernel(float4* __restrict__ p, int n4) {
    int i = blockIdx.x * blockDim.x + threadIdx.x;
    if (i < n4) p[i] = make_float4(0.f, 0.f, 0.f, 0.f);
}

// ---------------------------------------------------------------------------
// Kernel 3: per-edge message + scatter-add.  One edge per wave32, lane = ch m.
// Acc layout per node (256 floats): [s(64) | v0(64) | v1(64) | v2(64)]
// ---------------------------------------------------------------------------
__global__ void edge_kernel(const float* __restrict__ U,
                            const float* __restrict__ esh,
                            const float* __restrict__ rad,
                            const float* __restrict__ Wr1,
                            const float* __restrict__ Wr2,
                            const int* __restrict__ senders,
                            const int* __restrict__ receivers,
                            float* __restrict__ Acc) {
    const int e = rfl(blockIdx.x * 8 + (threadIdx.x >> 5));  // scalar edge id
    const int m = threadIdx.x & 31;

    const int snd = rfl(senders[e]);     // scalar
    const int rcv = rfl(receivers[e]);   // scalar

    const float* __restrict__ Ue = U + (size_t)snd * 128;
    const float ss  = Ue[m];
    const float vs0 = Ue[32 + m];
    const float vs1 = Ue[64 + m];
    const float vs2 = Ue[96 + m];

    const float4 y4 = *(const float4*)(esh + (size_t)e * 4);  // uniform
    const float y0 = y4.y, y1 = y4.z, y2 = y4.w;

    const float4 r0 = *(const float4*)(rad + (size_t)e * 8);      // uniform
    const float4 r1 = *(const float4*)(rad + (size_t)e * 8 + 4);
    const float r[8] = { r0.x, r0.y, r0.z, r0.w, r1.x, r1.y, r1.z, r1.w };

    const float inv8 = 0.35355339059327373f;   // 8^-0.5
    float h[8];
    #pragma unroll
    for (int j = 0; j < 8; ++j) {
        float a = 0.0f;
        #pragma unroll
        for (int k = 0; k < 8; ++k) a += r[k] * Wr1[k * 8 + j];
        h[j] = silu_f(a * inv8);
    }
    float w[4];                                // w_ss, w_vs, w_vt, w_st
    #pragma unroll
    for (int ci = 0; ci < 4; ++ci) {
        float a = 0.0f;
        #pragma unroll
        for (int j = 0; j < 8; ++j) a += h[j] * Wr2[j * 128 + ci * 32 + m];
        w[ci] = a * inv8;
    }

    const float mstp = (vs0 * y0 + vs1 * y1 + vs2 * y2) * 0.5773502691896258f;
    float* __restrict__ A = Acc + (size_t)rcv * 256;   // uniform base

    atomicAdd(A + m,        ss   * w[0]);   // m_s first half
    atomicAdd(A + 32 + m,   mstp * w[3]);   // m_s second half
    atomicAdd(A + 64 + m,   vs0 * w[1]);    // m_v first half
    atomicAdd(A + 128 + m,  vs1 * w[1]);
    atomicAdd(A + 192 + m,  vs2 * w[1]);
    const float t = ss * w[2];               // m_v second half
    atomicAdd(A + 96 + m,   t * y0);
    atomicAdd(A + 160 + m,  t * y1);
    atomicAdd(A + 224 + m,  t * y2);
}

// ---------------------------------------------------------------------------
// Kernel 4: down-projection.  G[n] (160 floats) = [g_s(64) | gv0 | gv1 | gv2]
// g_s  = (S * 16^-0.5) @ W_dn_s * 64^-0.5      (K=64, N=64)
// g_vi = (V_i * 16^-0.5) @ W_dn_v * 64^-0.5    (K=64, N=32)
// 10 waves per block; one unified loop parameterized by scalar uniforms.
// ---------------------------------------------------------------------------
__global__ void down_kernel(const float* __restrict__ Acc,
                            const float* __restrict__ Wds,
                            const float* __restrict__ Wdv,
                            float* __restrict__ G) {
    const int wave = rfl(threadIdx.x >> 5);   // 0..9  (scalar)
    const int lane = threadIdx.x & 31;
    const int half = lane >> 4;
    const int lm   = lane & 15;
    const int nbase = blockIdx.x * 16;

    // scalar per-wave parameters
    int aoff, bcol, ncols, goff;
    const float* __restrict__ B;
    if (wave < 4) {                       // g_s, N slice = wave
        B = Wds; ncols = 64; aoff = 0;
        bcol = wave * 16; goff = wave * 16;
    } else {                              // g_v component i, N slice nb
        const int i  = (wave - 4) >> 1;
        const int nb = (wave - 4) & 1;
        B = Wdv; ncols = 32; aoff = 64 + 64 * i;
        bcol = nb * 16; goff = 64 + 32 * i + nb * 16;
    }

    const float* __restrict__ Ap = Acc + (size_t)(nbase + lm) * 256
                                       + aoff + 2 * half;          // contiguous pair
    const float* __restrict__ Bp = B + (2 * half) * ncols + bcol + lm;

    v8f acc = {};
    #pragma unroll 4
    for (int kk = 0; kk < 16; ++kk) {
        v2f a = *(const v2f*)(Ap + 4 * kk);    // 8B aligned -> b64 load
        v2f b;
        b.x = Bp[(4 * kk) * ncols];
        b.y = Bp[(4 * kk + 1) * ncols];
        acc = __builtin_amdgcn_wmma_f32_16x16x4_f32(false, a, false, b,
                                                    (short)0, acc, false, false);
    }
    const float scale = 0.03125f;           // 16^-0.5 * 64^-0.5
    #pragma unroll
    for (int r = 0; r < 8; ++r) {
        const int row = nbase + r + 8 * half;
        G[(size_t)row * 160 + goff + lm] = acc[r] * scale;
    }
}

// ---------------------------------------------------------------------------
// Kernel 5: species skip (per-row B matrix -> wave-shuffle dot products),
// 0.5*(g + sk), silu gating, interleaved (m*3+i) vector output layout.
// One node per wave; 8 waves per block.
// ---------------------------------------------------------------------------
__global__ void finish_kernel(const float* __restrict__ nf,
                              const float* __restrict__ G,
                              const float* __restrict__ Wsks,
                              const float* __restrict__ Wskv,
                              const int* __restrict__ species,
                              float* __restrict__ out) {
    const int n = rfl(blockIdx.x * 8 + (threadIdx.x >> 5));  // scalar node id
    const int m = threadIdx.x & 31;

    const float sm = nf[(size_t)n * 128 + m];
    const float* __restrict__ vp = nf + (size_t)n * 128 + 32 + 3 * m;
    const float v0 = vp[0], v1 = vp[1], v2 = vp[2];

    const int sp = rfl(species[n]);                            // scalar
    const float* __restrict__ Ws = Wsks + (size_t)sp * 2048;   // [32][64]
    const float* __restrict__ Wv = Wskv + (size_t)sp * 1024;   // [32][32]

    float sk0 = 0.f, sk1 = 0.f, kv0 = 0.f, kv1 = 0.f, kv2 = 0.f;
    #pragma unroll 8
    for (int j = 0; j < 32; ++j) {
        const float sj  = __shfl(sm, j, 32);
        const float uj0 = __shfl(v0, j, 32);
        const float uj1 = __shfl(v1, j, 32);
        const float uj2 = __shfl(v2, j, 32);
        sk0 += sj * Ws[j * 64 + m];
        sk1 += sj * Ws[j * 64 + 32 + m];
        const float wv = Wv[j * 32 + m];
        kv0 += uj0 * wv; kv1 += uj1 * wv; kv2 += uj2 * wv;
    }
    const float inv32 = 0.17677669529663687f;  // 32^-0.5
    const float* __restrict__ Gp = G + (size_t)n * 160;
    const float gs0 = 0.5f * (Gp[m]        + sk0 * inv32);
    const float gs1 = 0.5f * (Gp[32 + m]   + sk1 * inv32);
    const float gv0 = 0.5f * (Gp[64 + m]   + kv0 * inv32);
    const float gv1 = 0.5f * (Gp[96 + m]   + kv1 * inv32);
    const float gv2 = 0.5f * (Gp[128 + m]  + kv2 * inv32);

    const float feat = silu_f(gs0);
    const float gate = silu_f(gs1);
    out[(size_t)n * 128 + m] = feat;
    float* __restrict__ op = out + (size_t)n * 128 + 32 + 3 * m;
    op[0] = gv0 * gate;
    op[1] = gv1 * gate;
    op[2] = gv2 * gate;
}

// ---------------------------------------------------------------------------
extern "C" void kernel_launch(void* const* d_in, const int* in_sizes, int n_in,
                              void* d_out, int out_size, void* d_ws, size_t ws_size,
                              hipStream_t stream) {
    const float* nf        = (const float*)d_in[0];
    const float* esh       = (const float*)d_in[1];
    const float* rad       = (const float*)d_in[2];
    const float* Wus       = (const float*)d_in[3];
    const float* Wuv       = (const float*)d_in[4];
    const float* Wr1       = (const float*)d_in[5];
    const float* Wr2       = (const float*)d_in[6];
    const float* Wds       = (const float*)d_in[7];
    const float* Wdv       = (const float*)d_in[8];
    const float* Wsks      = (const float*)d_in[9];
    const float* Wskv      = (const float*)d_in[10];
    const int*   senders   = (const int*)d_in[11];
    const int*   receivers = (const int*)d_in[12];
    const int*   species   = (const int*)d_in[13];
    float* out = (float*)d_out;

    float* U   = (float*)d_ws;                // N_NODES * 128
    float* Acc = U + (size_t)N_NODES * 128;   // N_NODES * 256
    float* G   = Acc + (size_t)N_NODES * 256; // N_NODES * 160

    // 1) up-projection (WMMA f32): 1024 node tiles, 8 waves each
    up_kernel<<<N_NODES / 16, 256, 0, stream>>>(nf, Wus, Wuv, U);

    // 2) zero accumulators (b128 stores)
    const int accN4 = N_NODES * 256 / 4;
    zero_kernel<<<(accN4 + 255) / 256, 256, 0, stream>>>((float4*)Acc, accN4);

    // 3) per-edge messages + atomic scatter (one edge per wave)
    edge_kernel<<<N_EDGES / 8, 256, 0, stream>>>(U, esh, rad, Wr1, Wr2,
                                                 senders, receivers, Acc);

    // 4) down-projection (WMMA f32): 1024 node tiles, 10 waves each
    down_kernel<<<N_NODES / 16, 320, 0, stream>>>(Acc, Wds, Wdv, G);

    // 5) skip + gate + output
    finish_kernel<<<N_NODES / 8, 256, 0, stream>>>(nf, G, Wsks, Wskv, species, out);
}